// NGCFLayer_19928648253535
// MI455X (gfx1250) — compile-verified
//
#include <hip/hip_runtime.h>

#define N_NODES 100000
#define N_EDGES 1200000
#define D 64
#define NROWBLK (N_NODES / 16)   // 6250 exact

typedef __attribute__((ext_vector_type(2))) float v2f;
typedef __attribute__((ext_vector_type(8))) float v8f;

// ---------------------------------------------------------------------------
// Kernel 1: zero the two accumulation buffers (f1 lives in d_out, f2 in d_ws)
// ---------------------------------------------------------------------------
__global__ void ngcf_zero(float* __restrict__ f1, float* __restrict__ f2) {
    int i = blockIdx.x * blockDim.x + threadIdx.x;           // one float4 each
    const int total = N_NODES * D / 4;
    if (i < total) {
        float4 z = make_float4(0.f, 0.f, 0.f, 0.f);
        ((float4*)f1)[i] = z;
        ((float4*)f2)[i] = z;
    }
}

// ---------------------------------------------------------------------------
// Kernel 2: per-edge gather + scaled message + atomic scatter-sum.
// 16 lanes cooperate on one edge; each lane handles one float4 chunk of D=64,
// so each x-row gather is a fully coalesced 256B transaction (L2-resident).
// ---------------------------------------------------------------------------
__global__ __launch_bounds__(256) void ngcf_edge(
    const float* __restrict__ x, const float* __restrict__ norm,
    const int* __restrict__ src, const int* __restrict__ dst,
    float* __restrict__ f1, float* __restrict__ f2) {
    int tid = blockIdx.x * 256 + threadIdx.x;
    int e = tid >> 4;
    int c = tid & 15;                     // float4 chunk within the row
    if (e >= N_EDGES) return;
    int s = src[e];
    int d = dst[e];
    float coef = norm[s] * norm[d];
    float4 xs = ((const float4*)(x + (size_t)s * D))[c];
    float4 xd = ((const float4*)(x + (size_t)d * D))[c];
    float4 m1, m2;
    m1.x = xs.x * coef;            m1.y = xs.y * coef;
    m1.z = xs.z * coef;            m1.w = xs.w * coef;
    m2.x = xs.x * xd.x * coef;     m2.y = xs.y * xd.y * coef;
    m2.z = xs.z * xd.z * coef;     m2.w = xs.w * xd.w * coef;
    float* p1 = f1 + (size_t)d * D + c * 4;
    float* p2 = f2 + (size_t)d * D + c * 4;
    atomicAdd(p1 + 0, m1.x); atomicAdd(p1 + 1, m1.y);
    atomicAdd(p1 + 2, m1.z); atomicAdd(p1 + 3, m1.w);
    atomicAdd(p2 + 0, m2.x); atomicAdd(p2 + 1, m2.y);
    atomicAdd(p2 + 2, m2.z); atomicAdd(p2 + 3, m2.w);
}

// ---------------------------------------------------------------------------
// Kernel 3: out = f1 @ W1 + f2 @ W2 + (b1 + b2), fp32 WMMA 16x16x4.
// 256 threads = 8 waves; each wave produces one 16-row x 64-col output tile.
// Weights are staged in LDS interleaved by K-pairs:
//   Wp[(k/2)*64 + n] = { W[k][n], W[k+1][n] }   (k even)
// so each B fragment (needs K=kb,kb+1 for its column) is ONE aligned
// ds_load_b64 straight into the even VGPR pair WMMA consumes — no repacking.
// f1 aliases `out` (d_out): a wave reads only its own 16 rows before storing
// them, and no other wave touches those rows -> safe.
// ---------------------------------------------------------------------------
__global__ __launch_bounds__(256) void ngcf_gemm(
    const float* __restrict__ f1, const float* __restrict__ f2,
    const float* __restrict__ W1, const float* __restrict__ W2,
    const float* __restrict__ b1, const float* __restrict__ b2,
    float* __restrict__ out) {
    __shared__ v2f W1p[D * D / 2];       // 16 KB: K-pair-interleaved W1
    __shared__ v2f W2p[D * D / 2];       // 16 KB: K-pair-interleaved W2
    int t = threadIdx.x;
    // Cooperative interleaved fill: i -> (kpair = i/64, n = i%64).
    // Consecutive threads hit consecutive n => both global b32 loads coalesce.
    for (int i = t; i < D * D / 2; i += 256) {
        int kp = i >> 6;                 // k/2
        int n  = i & 63;
        v2f w1, w2;
        w1.x = W1[(2 * kp) * D + n];  w1.y = W1[(2 * kp + 1) * D + n];
        w2.x = W2[(2 * kp) * D + n];  w2.y = W2[(2 * kp + 1) * D + n];
        W1p[i] = w1;
        W2p[i] = w2;
    }
    __syncthreads();

    int wave = t >> 5;
    int lane = t & 31;
    int rowBlk = blockIdx.x * 8 + wave;
    if (rowBlk >= NROWBLK) return;       // whole wave exits; EXEC stays full
    int row0 = rowBlk * 16;

    int lhalf = lane & 15;               // M (for A) / N (for B,C,D)
    int kb = (lane >> 4) * 2;            // K sub-offset: lanes 16..31 hold K+2

    // Per-lane bias for each N-tile column.
    float bias[4];
#pragma unroll
    for (int nt = 0; nt < 4; ++nt)
        bias[nt] = b1[nt * 16 + lhalf] + b2[nt * 16 + lhalf];

    v8f acc[4] = {};                     // 4 N-tiles of the 16x64 output
    const float* a1p = f1 + (size_t)(row0 + lhalf) * D;
    const float* a2p = f2 + (size_t)(row0 + lhalf) * D;

#pragma unroll
    for (int k0 = 0; k0 < D; k0 += 4) {
        // A fragments (16x4 f32): VGPR0 = K=kb, VGPR1 = K=kb+1 (one b64 load).
        v2f aA = *(const v2f*)(a1p + k0 + kb);
        v2f aB = *(const v2f*)(a2p + k0 + kb);
        int kpRow = ((k0 + kb) >> 1) * D;        // (k/2)*64, k0+kb is even
#pragma unroll
        for (int nt = 0; nt < 4; ++nt) {
            v2f b = W1p[kpRow + nt * 16 + lhalf];        // one ds_load_b64
            acc[nt] = __builtin_amdgcn_wmma_f32_16x16x4_f32(
                false, aA, false, b, (short)0, acc[nt], false, false);
        }
#pragma unroll
        for (int nt = 0; nt < 4; ++nt) {
            v2f b = W2p[kpRow + nt * 16 + lhalf];        // one ds_load_b64
            acc[nt] = __builtin_amdgcn_wmma_f32_16x16x4_f32(
                false, aB, false, b, (short)0, acc[nt], false, false);
        }
    }

    // C/D layout: VGPR i -> M = i + (lane>=16 ? 8 : 0), N = lane&15.
    int mbase = (lane >> 4) * 8;
#pragma unroll
    for (int nt = 0; nt < 4; ++nt) {
        int col = nt * 16 + lhalf;
#pragma unroll
        for (int i = 0; i < 8; ++i) {
            out[(size_t)(row0 + mbase + i) * D + col] = acc[nt][i] + bias[nt];
        }
    }
}

// ---------------------------------------------------------------------------
extern "C" void kernel_launch(void* const* d_in, const int* in_sizes, int n_in,
                              void* d_out, int out_size, void* d_ws, size_t ws_size,
                              hipStream_t stream) {
    const float* x    = (const float*)d_in[0];
    const float* norm = (const float*)d_in[1];
    const int*   src  = (const int*)d_in[2];
    const int*   dst  = (const int*)d_in[3];
    const float* W1   = (const float*)d_in[4];
    const float* b1   = (const float*)d_in[5];
    const float* W2   = (const float*)d_in[6];
    const float* b2   = (const float*)d_in[7];

    float* out = (float*)d_out;
    float* f1  = (float*)d_out;          // accumulate f1 in the output buffer
    float* f2  = (float*)d_ws;           // 25.6 MB scratch for f2

    // 1) zero accumulators
    {
        int total = N_NODES * D / 4;     // 1,600,000 float4 slots
        int blocks = (total + 255) / 256;
        ngcf_zero<<<blocks, 256, 0, stream>>>(f1, f2);
    }
    // 2) edge scatter (16 lanes per edge)
    {
        long long threads = (long long)N_EDGES * 16;
        int blocks = (int)((threads + 255) / 256);
        ngcf_edge<<<blocks, 256, 0, stream>>>(x, norm, src, dst, f1, f2);
    }
    // 3) WMMA combine
    {
        int blocks = (NROWBLK + 7) / 8;  // 8 waves (16-row tiles) per block
        ngcf_gemm<<<blocks, 256, 0, stream>>>(f1, f2, W1, W2, b1, b2, out);
    }
}